// GCN_68401649156307
// MI455X (gfx1250) — compile-verified
//
#include <hip/hip_runtime.h>
#include <hip/hip_bf16.h>

// ---------------------------------------------------------------------------
// 2-layer GCN for gfx1250 (MI455X).
//   GEMMs:    V_WMMA_F32_16X16X4_F32 (full fp32 precision, matches reference)
//   Scatter:  global_atomic_add_f32 into L2-resident agg buffers
// Problem constants from the reference: D_IN=128, D_HID=64, D_OUT=32.
// N (=100000) is a multiple of 16 -> clean WMMA tiling, no tail.
// ---------------------------------------------------------------------------

typedef float v2f __attribute__((ext_vector_type(2)));
typedef float v8f __attribute__((ext_vector_type(8)));

#define D_IN  128
#define D_HID 64
#define D_OUT 32

// ---------------- degree / norm ----------------

__global__ void k_zero_u32(unsigned* p, int n) {
    int i = blockIdx.x * blockDim.x + threadIdx.x;
    if (i < n) p[i] = 0u;
}

__global__ void k_degree(const int* __restrict__ ei, unsigned* __restrict__ deg, int E) {
    int e = blockIdx.x * blockDim.x + threadIdx.x;
    if (e < E) {
        int d = ei[E + e];            // edge_index[1][e] (row-major [2,E])
        atomicAdd(&deg[d], 1u);
    }
}

__global__ void k_dinv(const unsigned* __restrict__ deg, float* __restrict__ dinv, int N) {
    int i = blockIdx.x * blockDim.x + threadIdx.x;
    if (i < N) {
        float dg = (float)(deg[i] + 1u);   // +1 self-loop
        dinv[i] = rsqrtf(dg);              // deg >= 1 always
    }
}

// ---------------- WMMA GEMM 1: H = X @ W1   (K=128, Nout=64) ----------------
// One wave computes a 16x16 tile of H. blockDim = 128 (4 waves -> 4 N-tiles).
// fp32 WMMA layouts (ISA 7.12.2):
//   A 16x4 : lane l holds A[l%16][2*(l/16) + {0,1}] in v[0],v[1]
//   B 4x16 : lane l holds B[2*(l/16) + {0,1}][l%16]
//   C 16x16: VGPR r, lane l -> C[r + 8*(l/16)][l%16]

__global__ void k_gemm1_wmma(const float* __restrict__ X, const float* __restrict__ W,
                             float* __restrict__ H) {
    const int lane = threadIdx.x & 31;
    const int wave = threadIdx.x >> 5;      // 0..3
    const int m0   = blockIdx.x * 16;
    const int n0   = wave * 16;
    const int half = lane >> 4;             // 0 or 1
    const int lm   = lane & 15;

    const float* xrow = X + (size_t)(m0 + lm) * D_IN;
    v8f acc = {};
    #pragma unroll 4
    for (int k0 = 0; k0 < D_IN; k0 += 4) {
        const int ka = k0 + 2 * half;
        v2f a, b;
        a.x = xrow[ka];
        a.y = xrow[ka + 1];
        b.x = W[(size_t)ka       * D_HID + n0 + lm];
        b.y = W[(size_t)(ka + 1) * D_HID + n0 + lm];
        acc = __builtin_amdgcn_wmma_f32_16x16x4_f32(false, a, false, b,
                                                    (short)0, acc, false, false);
    }
    #pragma unroll
    for (int r = 0; r < 8; ++r) {
        const int m = m0 + r + 8 * half;
        H[(size_t)m * D_HID + n0 + lm] = acc[r];
    }
}

// ---- self-loop init: agg1 = dinv[n]^2 * h  (full overwrite, no pre-zero) ----
__global__ void k_selfinit64(const float* __restrict__ h, const float* __restrict__ dinv,
                             float* __restrict__ agg, int N) {
    int i = blockIdx.x * blockDim.x + threadIdx.x;   // over N*64
    if (i < N * D_HID) {
        int n = i >> 6;
        float w = dinv[n];
        agg[i] = w * w * h[i];
    }
}

// ---- edge scatter, layer 1: 4 edges x 64 features per 256-thread block ----
__global__ void k_edge64(const int* __restrict__ ei, const float* __restrict__ dinv,
                         const float* __restrict__ h, float* __restrict__ agg, int E) {
    int t = blockIdx.x * 256 + threadIdx.x;
    int e = t >> 6;
    int f = t & 63;
    if (e < E) {
        int s = ei[e];
        int d = ei[E + e];
        float w = dinv[s] * dinv[d];
        atomicAdd(&agg[(size_t)d * D_HID + f], h[(size_t)s * D_HID + f] * w);
    }
}

// ---------------- WMMA GEMM 2: H2 = relu(AGG + b1) @ W2  (K=64, Nout=32) ----
// blockDim = 64 (2 waves -> 2 N-tiles). relu+bias fused into the A-load.

__global__ void k_gemm2_wmma(const float* __restrict__ AGG, const float* __restrict__ b1,
                             const float* __restrict__ W, float* __restrict__ H2) {
    const int lane = threadIdx.x & 31;
    const int wave = threadIdx.x >> 5;      // 0..1
    const int m0   = blockIdx.x * 16;
    const int n0   = wave * 16;
    const int half = lane >> 4;
    const int lm   = lane & 15;

    const float* arow = AGG + (size_t)(m0 + lm) * D_HID;
    v8f acc = {};
    #pragma unroll 4
    for (int k0 = 0; k0 < D_HID; k0 += 4) {
        const int ka = k0 + 2 * half;
        v2f a, b;
        a.x = fmaxf(arow[ka]     + b1[ka],     0.f);
        a.y = fmaxf(arow[ka + 1] + b1[ka + 1], 0.f);
        b.x = W[(size_t)ka       * D_OUT + n0 + lm];
        b.y = W[(size_t)(ka + 1) * D_OUT + n0 + lm];
        acc = __builtin_amdgcn_wmma_f32_16x16x4_f32(false, a, false, b,
                                                    (short)0, acc, false, false);
    }
    #pragma unroll
    for (int r = 0; r < 8; ++r) {
        const int m = m0 + r + 8 * half;
        H2[(size_t)m * D_OUT + n0 + lm] = acc[r];
    }
}

// ---- out init: out = b2 + dinv^2 * h2 (full overwrite) ----
__global__ void k_outinit32(const float* __restrict__ h2, const float* __restrict__ dinv,
                            const float* __restrict__ b2, float* __restrict__ out, int N) {
    int i = blockIdx.x * blockDim.x + threadIdx.x;   // over N*32
    if (i < N * D_OUT) {
        int n = i >> 5;
        int f = i & 31;
        float w = dinv[n];
        out[i] = b2[f] + w * w * h2[i];
    }
}

// ---- edge scatter, layer 2: 8 edges x 32 features per 256-thread block ----
__global__ void k_edge32(const int* __restrict__ ei, const float* __restrict__ dinv,
                         const float* __restrict__ h2, float* __restrict__ out, int E) {
    int t = blockIdx.x * 256 + threadIdx.x;
    int e = t >> 5;
    int f = t & 31;
    if (e < E) {
        int s = ei[e];
        int d = ei[E + e];
        float w = dinv[s] * dinv[d];
        atomicAdd(&out[(size_t)d * D_OUT + f], h2[(size_t)s * D_OUT + f] * w);
    }
}

// ---------------------------------------------------------------------------

extern "C" void kernel_launch(void* const* d_in, const int* in_sizes, int n_in,
                              void* d_out, int out_size, void* d_ws, size_t ws_size,
                              hipStream_t stream) {
    const float* x   = (const float*)d_in[0];   // [N,128]
    const int*   ei  = (const int*)  d_in[1];   // [2,E]
    const float* W1  = (const float*)d_in[2];   // [128,64]
    const float* b1  = (const float*)d_in[3];   // [64]
    const float* W2  = (const float*)d_in[4];   // [64,32]
    const float* b2  = (const float*)d_in[5];   // [32]
    float*       out = (float*)d_out;           // [N,32]

    const int N = in_sizes[0] / D_IN;           // 100000
    const int E = in_sizes[1] / 2;              // 1600000

    // Workspace carve-up (bytes): deg | dinv | h | agg1 | h2
    char* ws = (char*)d_ws;
    unsigned* deg  = (unsigned*)(ws);
    float*    dinv = (float*)(ws + (size_t)N * 4);
    float*    h    = (float*)(ws + (size_t)N * 8);
    float*    agg1 = (float*)(ws + (size_t)N * 8 + (size_t)N * D_HID * 4);
    float*    h2   = (float*)(ws + (size_t)N * 8 + (size_t)N * D_HID * 8);

    const int TB = 256;

    // 1) degree (with self-loop folded in at rsqrt)
    k_zero_u32<<<(N + TB - 1) / TB, TB, 0, stream>>>(deg, N);
    k_degree  <<<(E + TB - 1) / TB, TB, 0, stream>>>(ei, deg, E);
    k_dinv    <<<(N + TB - 1) / TB, TB, 0, stream>>>(deg, dinv, N);

    // 2) layer 1: h = X@W1 ; agg1 = dinv^2*h + scatter(edges)
    k_gemm1_wmma<<<N / 16, 128, 0, stream>>>(x, W1, h);
    k_selfinit64<<<((size_t)N * D_HID + TB - 1) / TB, TB, 0, stream>>>(h, dinv, agg1, N);
    k_edge64    <<<((size_t)E * D_HID + TB - 1) / TB, TB, 0, stream>>>(ei, dinv, h, agg1, E);

    // 3) layer 2: h2 = relu(agg1+b1)@W2 ; out = b2 + dinv^2*h2 + scatter(edges)
    k_gemm2_wmma<<<N / 16, 64, 0, stream>>>(agg1, b1, W2, h2);
    k_outinit32 <<<((size_t)N * D_OUT + TB - 1) / TB, TB, 0, stream>>>(h2, dinv, b2, out, N);
    k_edge32    <<<((size_t)E * D_OUT + TB - 1) / TB, TB, 0, stream>>>(ei, dinv, h2, out, E);
}